// EfficientDet_14396730376686
// MI455X (gfx1250) — compile-verified
//
#include <hip/hip_runtime.h>
#include <hip/hip_bf16.h>

// ---------------------------------------------------------------------------
// EfficientDet post-process: score-max -> radix top-k -> sort -> decode ->
// IoU (WMMA union outer-sum) -> greedy NMS -> gather outputs.
// ---------------------------------------------------------------------------

typedef __attribute__((ext_vector_type(2))) float v2f;
typedef __attribute__((ext_vector_type(8))) float v8f;

#define KKEEP 1000
#define NPAD  1024
#define THRESH 0.5f
#define IOU_THRESH 0.5f

__device__ __forceinline__ unsigned fkey(float f) {
    unsigned b = __float_as_uint(f);
    return (b & 0x80000000u) ? ~b : (b | 0x80000000u); // monotone: bigger float -> bigger key
}

// ---------------------------------------------------------------------------
// Kernel 1: per-anchor max over C=80 classes. 141 MB stream -> bandwidth bound.
// Stage cla rows through LDS with CDNA5 async global->LDS copies (ASYNCcnt).
// 128 anchors/block, 80 floats each = 40 KB LDS tile (8 blocks/WGP occupancy).
// ---------------------------------------------------------------------------
__global__ __launch_bounds__(128) void score_max_kernel(
    const float* __restrict__ cla, float* __restrict__ scores, int A, int C) {
    __shared__ float4 tile[128 * 20]; // 40 KB, at LDS offset 0 (only shared var)
    const int tid = threadIdx.x;
    const long long base = (long long)blockIdx.x * 128;
    const int cnt = min(128, (int)(A - base));
    const int cq  = C >> 2;           // 20 float4 per anchor
    const int nf4 = cnt * cq;
    const float4* src = (const float4*)(cla + base * (long long)C);

    for (int i = tid; i < nf4; i += 128) {
        unsigned lds_off = (unsigned)(i * 16);
        unsigned long long ga = (unsigned long long)(const void*)(src + i);
        asm volatile("global_load_async_to_lds_b128 %0, %1, off"
                     :: "v"(lds_off), "v"(ga) : "memory");
    }
    asm volatile("s_wait_asynccnt 0x0" ::: "memory");
    __syncthreads();

    if (tid < cnt) {
        const int q0 = tid * cq;
        float4 v = tile[q0];
        float mv = fmaxf(fmaxf(v.x, v.y), fmaxf(v.z, v.w));
        for (int q = 1; q < cq; ++q) {
            v = tile[q0 + q];
            mv = fmaxf(mv, fmaxf(fmaxf(v.x, v.y), fmaxf(v.z, v.w)));
        }
        scores[base + tid] = mv;
    }
}

// ---------------------------------------------------------------------------
// Radix-select plumbing: state[0]=prefix/threshold key, state[1]=k remaining,
// state[2]=count(> T), state[3]=count(== T)
// ---------------------------------------------------------------------------
__global__ void init_kernel(unsigned* __restrict__ state, unsigned* __restrict__ hist,
                            int* __restrict__ cidx, float* __restrict__ cscore) {
    int t = blockIdx.x * blockDim.x + threadIdx.x;
    if (t < 256) hist[t] = 0u;
    if (t == 0) { state[0] = 0u; state[1] = (unsigned)KKEEP; state[2] = 0u; state[3] = 0u; }
    if (t < NPAD) { cidx[t] = -1; cscore[t] = -1e30f; }
}

__global__ __launch_bounds__(256) void radix_hist_kernel(
    const float* __restrict__ scores, int A, unsigned* __restrict__ hist,
    const unsigned* __restrict__ state, int pass) {
    __shared__ unsigned lh[256];
    const int t = threadIdx.x;
    lh[t] = 0u;
    __syncthreads();
    const unsigned prefix = state[0];
    const int shift = 24 - 8 * pass;
    for (int i = blockIdx.x * 256 + t; i < A; i += gridDim.x * 256) {
        unsigned k = fkey(scores[i]);
        bool ok = (pass == 0) || ((k >> (shift + 8)) == prefix);
        if (ok) atomicAdd(&lh[(k >> shift) & 255u], 1u);
    }
    __syncthreads();
    if (lh[t]) atomicAdd(&hist[t], lh[t]);
}

__global__ __launch_bounds__(256) void radix_scan_kernel(
    unsigned* __restrict__ hist, unsigned* __restrict__ state) {
    __shared__ unsigned h[256];
    const int t = threadIdx.x;
    h[t] = hist[t];
    __syncthreads();
    if (t == 0) {
        unsigned need = state[1];
        unsigned prefix = state[0];
        int d = 255;
        for (; d > 0; --d) {
            unsigned c = h[d];
            if (need <= c) break;
            need -= c;
        }
        state[0] = (prefix << 8) | (unsigned)d;
        state[1] = need;
    }
    __syncthreads();
    hist[t] = 0u; // ready for next pass
}

__global__ __launch_bounds__(256) void select_kernel(
    const float* __restrict__ scores, int A, unsigned* __restrict__ state,
    int* __restrict__ cidx, float* __restrict__ cscore) {
    const unsigned T = state[0];
    const unsigned kneed = state[1];
    const int G = KKEEP - (int)kneed; // exact count of keys strictly greater than T
    for (int i = blockIdx.x * 256 + threadIdx.x; i < A; i += gridDim.x * 256) {
        unsigned k = fkey(scores[i]);
        if (k > T) {
            unsigned p = atomicAdd(&state[2], 1u);
            cidx[p] = i; cscore[p] = scores[i];
        } else if (k == T) {
            unsigned e = atomicAdd(&state[3], 1u);
            if (e < kneed) { int p = G + (int)e; cidx[p] = i; cscore[p] = scores[i]; }
        }
    }
}

// ---------------------------------------------------------------------------
// Bitonic sort of 1024 (score desc, idx asc) pairs in one workgroup.
// ---------------------------------------------------------------------------
__global__ __launch_bounds__(1024) void sort_kernel(int* __restrict__ cidx,
                                                    float* __restrict__ cscore) {
    __shared__ float sk[NPAD];
    __shared__ int   si[NPAD];
    const int t = threadIdx.x;
    sk[t] = cscore[t];
    si[t] = cidx[t];
    for (int k = 2; k <= NPAD; k <<= 1) {
        for (int j = k >> 1; j > 0; j >>= 1) {
            __syncthreads();
            int x = t ^ j;
            if (x > t) {
                float s1 = sk[t], s2 = sk[x];
                int   i1 = si[t], i2 = si[x];
                bool b12 = (s1 > s2) || (s1 == s2 && i1 < i2); // "1 before 2" descending
                bool desc = ((t & k) == 0);
                bool sw = desc ? !b12 : b12;
                if (sw) { sk[t] = s2; si[t] = i2; sk[x] = s1; si[x] = i1; }
            }
        }
    }
    __syncthreads();
    cscore[t] = sk[t];
    cidx[t]   = si[t];
}

// ---------------------------------------------------------------------------
// Decode + clip the 1000 candidate boxes.
// ---------------------------------------------------------------------------
__global__ __launch_bounds__(1024) void decode_kernel(
    const float4* __restrict__ reg, const float4* __restrict__ anchors,
    const int* __restrict__ cidx, const int* __restrict__ hptr,
    const int* __restrict__ wptr, float4* __restrict__ cbox) {
    const int t = threadIdx.x;
    const int a = cidx[t];
    float4 o = {0.f, 0.f, 0.f, 0.f};
    if (a >= 0) {
        float4 an = anchors[a];
        float4 dg = reg[a];
        float W = (float)(*wptr) - 1.0f;
        float H = (float)(*hptr) - 1.0f;
        float wa = an.z - an.x, ha = an.w - an.y;
        float cxa = an.x + 0.5f * wa, cya = an.y + 0.5f * ha;
        float cx = cxa + dg.x * wa, cy = cya + dg.y * ha;
        float w = wa * expf(dg.z), h = ha * expf(dg.w);
        o.x = fminf(fmaxf(cx - 0.5f * w, 0.f), W);
        o.y = fminf(fmaxf(cy - 0.5f * h, 0.f), H);
        o.z = fminf(fmaxf(cx + 0.5f * w, 0.f), W);
        o.w = fminf(fmaxf(cy + 0.5f * h, 0.f), H);
    }
    cbox[t] = o;
}

// ---------------------------------------------------------------------------
// IoU matrix 1024x1024 in 16x16 tiles, one tile per wave32.
// Union base area_i[m] + area_j[n] is an outer sum == rank-2 matmul:
//   A(16x4) = [area_i | 1 | 0 | 0],  B(4x16) = [1 ; area_j ; 0 ; 0]
// computed with V_WMMA_F32_16X16X4_F32; intersection fused per element.
// Division replaced by single v_rcp_f32 (IoU is only compared vs 0.5).
// ---------------------------------------------------------------------------
__global__ __launch_bounds__(128) void iou_kernel(const float4* __restrict__ box,
                                                  float* __restrict__ iou) {
    const int wave = threadIdx.x >> 5;
    const int lane = threadIdx.x & 31;
    const int tile = blockIdx.x * 4 + wave;   // 0..4095
    const int m0 = (tile >> 6) * 16;
    const int n0 = (tile & 63) * 16;
    const int half = lane >> 4;
    const int ln   = lane & 15;

    const int n = n0 + ln;
    const float4 bj = box[n];
    const float areaj = fmaxf(bj.z - bj.x, 0.f) * fmaxf(bj.w - bj.y, 0.f);

    const float4 bi0 = box[m0 + ln];
    const float areai = fmaxf(bi0.z - bi0.x, 0.f) * fmaxf(bi0.w - bi0.y, 0.f);

    const float z = (half == 0) ? 1.0f : 0.0f; // zero the K>=2 half of the wave
    v2f a; a.x = areai * z; a.y = z;          // A: K0=area_i, K1=1
    v2f b; b.x = z;         b.y = areaj * z;  // B: K0=1,      K1=area_j
    v8f c = {};
#if __has_builtin(__builtin_amdgcn_wmma_f32_16x16x4_f32)
    v8f d = __builtin_amdgcn_wmma_f32_16x16x4_f32(false, a, false, b,
                                                  (short)0, c, false, false);
#else
    v8f d = c;
    for (int r = 0; r < 8; ++r) {
        float4 bi = box[m0 + r + 8 * half];
        d[r] = fmaxf(bi.z - bi.x, 0.f) * fmaxf(bi.w - bi.y, 0.f) + areaj;
    }
#endif
    for (int r = 0; r < 8; ++r) {
        const int m = m0 + r + 8 * half;      // C/D layout: VGPR r -> rows r / r+8
        const float4 bi = box[m];
        float xx1 = fmaxf(bi.x, bj.x), yy1 = fmaxf(bi.y, bj.y);
        float xx2 = fminf(bi.z, bj.z), yy2 = fminf(bi.w, bj.w);
        float inter = fmaxf(xx2 - xx1, 0.f) * fmaxf(yy2 - yy1, 0.f);
        float uni = d[r] - inter;             // area_i + area_j - inter (from WMMA)
        // v_rcp_f32 instead of IEEE div expansion: ~1 ulp, threshold-compare only
        float iouv = inter * __builtin_amdgcn_rcpf(fmaxf(uni, 1e-8f));
        iou[(size_t)m * NPAD + n] = iouv;
    }
}

// ---------------------------------------------------------------------------
// Greedy sequential NMS: one 1024-thread workgroup, keep flags in LDS.
// Next row's IoU value is prefetched across the barrier to hide L2 latency.
// ---------------------------------------------------------------------------
__global__ __launch_bounds__(1024) void nms_kernel(const float* __restrict__ cscore,
                                                   const float* __restrict__ iou,
                                                   int* __restrict__ keepg) {
    __shared__ unsigned char keep[NPAD];
    __shared__ int sup;
    const int t = threadIdx.x;
    const float sc = cscore[t];
    const bool valid = (t < KKEEP) && (sc > THRESH);
    keep[t] = 0;
    __syncthreads();
    float v = iou[t]; // row 0
    for (int i = 0; i < KKEEP; ++i) {
        float vn = (i + 1 < KKEEP) ? iou[(size_t)(i + 1) * NPAD + t] : 0.f;
        if (t == 0) sup = 0;
        __syncthreads();
        if (keep[t] && v > IOU_THRESH) sup = 1; // benign race, all write 1
        __syncthreads();
        if (t == i) keep[i] = (valid && sup == 0) ? 1 : 0;
        __syncthreads();
        v = vn;
    }
    keepg[t] = keep[t];
}

// ---------------------------------------------------------------------------
// Final gather: scores, argmax class (first max), boxes.
// out layout: [1000 scores][1000 class-as-float][1000*4 boxes]
// ---------------------------------------------------------------------------
__global__ __launch_bounds__(128) void output_kernel(
    const float* __restrict__ cla, const int* __restrict__ cidx,
    const float4* __restrict__ cbox, const int* __restrict__ keep,
    float* __restrict__ out, int C) {
    const int i = blockIdx.x * blockDim.x + threadIdx.x;
    if (i >= KKEEP) return;
    float sc = 0.f, cl = -1.f;
    float4 bx = {0.f, 0.f, 0.f, 0.f};
    if (keep[i]) {
        const int a = cidx[i];
        const float* row = cla + (size_t)a * C;
        float mv = row[0];
        int arg = 0;
        for (int c2 = 1; c2 < C; ++c2) {
            float vv = row[c2];
            if (vv > mv) { mv = vv; arg = c2; }
        }
        sc = mv; cl = (float)arg; bx = cbox[i];
    }
    out[i] = sc;
    out[KKEEP + i] = cl;
    ((float4*)(out + 2 * KKEEP))[i] = bx;
}

// ---------------------------------------------------------------------------
extern "C" void kernel_launch(void* const* d_in, const int* in_sizes, int n_in,
                              void* d_out, int out_size, void* d_ws, size_t ws_size,
                              hipStream_t stream) {
    const float* cla     = (const float*)d_in[0];
    const float* reg     = (const float*)d_in[1];
    const float* anchors = (const float*)d_in[2];
    const int*   hptr    = (const int*)d_in[3];
    const int*   wptr    = (const int*)d_in[4];

    const int A = in_sizes[1] / 4;         // 441936
    const int C = in_sizes[0] / A;         // 80

    // workspace carve-up (256B aligned)
    char* w = (char*)d_ws;
    size_t off = 0;
    auto carve = [&](size_t bytes) -> char* {
        char* p = w + off;
        off = (off + bytes + 255) & ~(size_t)255;
        return p;
    };
    float*    scores = (float*)   carve((size_t)A * 4);
    unsigned* state  = (unsigned*)carve(64 * 4);
    unsigned* hist   = (unsigned*)carve(256 * 4);
    int*      cidx   = (int*)     carve(NPAD * 4);
    float*    cscore = (float*)   carve(NPAD * 4);
    float4*   cbox   = (float4*)  carve(NPAD * 16);
    int*      keepg  = (int*)     carve(NPAD * 4);
    float*    iou    = (float*)   carve((size_t)NPAD * NPAD * 4);
    (void)ws_size; (void)n_in; (void)out_size;

    // 1) per-anchor class max (async global->LDS streaming)
    score_max_kernel<<<(A + 127) / 128, 128, 0, stream>>>(cla, scores, A, C);

    // 2) radix top-1000 selection
    init_kernel<<<4, 256, 0, stream>>>(state, hist, cidx, cscore);
    for (int pass = 0; pass < 4; ++pass) {
        radix_hist_kernel<<<512, 256, 0, stream>>>(scores, A, hist, state, pass);
        radix_scan_kernel<<<1, 256, 0, stream>>>(hist, state);
    }
    select_kernel<<<512, 256, 0, stream>>>(scores, A, state, cidx, cscore);

    // 3) sort candidates (score desc, idx asc) to match top_k order
    sort_kernel<<<1, 1024, 0, stream>>>(cidx, cscore);

    // 4) decode + clip candidate boxes
    decode_kernel<<<1, 1024, 0, stream>>>((const float4*)reg, (const float4*)anchors,
                                          cidx, hptr, wptr, cbox);

    // 5) IoU matrix via WMMA outer-sum union tiles
    iou_kernel<<<1024, 128, 0, stream>>>(cbox, iou);

    // 6) greedy NMS
    nms_kernel<<<1, 1024, 0, stream>>>(cscore, iou, keepg);

    // 7) outputs
    output_kernel<<<(KKEEP + 127) / 128, 128, 0, stream>>>(cla, cidx, cbox, keepg,
                                                           (float*)d_out, C);
}